// DenseEquExplicit_7241314861254
// MI455X (gfx1250) — compile-verified
//
#include <hip/hip_runtime.h>
#include <cstdint>
#include <cstddef>

// Equirect bicubic grid-sample + sigmoid for MI455X (gfx1250).
// Gather-bound: texture (~101 MB) is L2-resident (192 MB L2). Per point we do
// 12 unaligned b128 gathers (4 contiguous x-taps x 3 channels x 4 rows),
// stage the streamed xyz input through LDS with CDNA5 async DMA (ASYNCcnt),
// and keep transcendental VALU lean (rsq/rcp/exp fast intrinsics).

#define BLK 256
#define GH 2048
#define GW 4104                      // EQU_W + 2*PAD
#define GPLANE ((size_t)GH * (size_t)GW)

typedef int v4i_t __attribute__((ext_vector_type(4)));
typedef float f4v __attribute__((ext_vector_type(4)));
typedef __attribute__((address_space(1))) v4i_t* g_v4i_ptr;   // global
typedef __attribute__((address_space(3))) v4i_t* l_v4i_ptr;   // LDS

#if defined(__has_builtin)
#if __has_builtin(__builtin_amdgcn_global_load_async_to_lds_b128)
#define HAVE_ASYNC_B128_BUILTIN 1
#endif
#if __has_builtin(__builtin_amdgcn_s_wait_asynccnt)
#define HAVE_WAIT_ASYNC_BUILTIN 1
#endif
#endif

__device__ __forceinline__ void lds_async_copy16(const float* g, float* l) {
#if defined(HAVE_ASYNC_B128_BUILTIN)
  void* gv = (void*)g;  // strip const, then addrspace-cast
  void* lv = (void*)l;
  __builtin_amdgcn_global_load_async_to_lds_b128(
      (g_v4i_ptr)gv, (l_v4i_ptr)lv, 0, 0);
#else
  unsigned loff = (unsigned)(uintptr_t)(__attribute__((address_space(3))) float*)l;
  asm volatile("global_load_async_to_lds_b128 %0, %1, off"
               :
               : "v"(loff), "v"(g)
               : "memory");
#endif
}

__device__ __forceinline__ void lds_async_wait0() {
#if defined(HAVE_WAIT_ASYNC_BUILTIN)
  __builtin_amdgcn_s_wait_asynccnt(0);
#else
  asm volatile("s_wait_asynccnt 0" ::: "memory");
#endif
}

// PyTorch bicubic convolution weights, a = -0.75, taps at offsets -1..+2.
__device__ __forceinline__ void cubic_w(float t, float w[4]) {
  const float a = -0.75f;
  const float t1 = t + 1.0f;
  const float t4 = 2.0f - t;
  const float s = 1.0f - t;
  w[0] = ((a * t1 - 5.0f * a) * t1 + 8.0f * a) * t1 - 4.0f * a;
  w[1] = ((a + 2.0f) * t - (a + 3.0f)) * t * t + 1.0f;
  w[2] = ((a + 2.0f) * s - (a + 3.0f)) * s * s + 1.0f;
  w[3] = ((a * t4 - 5.0f * a) * t4 + 8.0f * a) * t4 - 4.0f * a;
}

__device__ __forceinline__ f4v load_f4_unaligned(const float* p) {
  f4v v;
  __builtin_memcpy(&v, p, sizeof(f4v));  // align-4: b128 in UNALIGNED mode
  return v;
}

__device__ __forceinline__ float sigmoid_fast(float x) {
  // 1/(1+exp(-x)) with v_exp_f32 + v_rcp_f32
  return __builtin_amdgcn_rcpf(1.0f + __expf(-x));
}

__global__ __launch_bounds__(BLK) void equ_bicubic_sample(
    const float* __restrict__ xyz, const float* __restrict__ grid,
    float* __restrict__ out, int npts) {
  __shared__ float sxyz[BLK * 3];
  const int tid = (int)threadIdx.x;
  const long long base = (long long)blockIdx.x * (BLK * 3);
  const long long total = (long long)npts * 3;

  // ---- Stage this block's xyz slice (768 floats) into LDS via async DMA ----
  // 192 lanes x b128 (16B, aligned: block granularity is 3072 B).
  if (tid < (BLK * 3) / 4) {
    const long long fofs = base + (long long)tid * 4;
    if (fofs + 4 <= total) {
      lds_async_copy16(xyz + fofs, &sxyz[tid * 4]);
    } else if (fofs < total) {           // ragged tail (not hit for N=2^21)
      for (long long f = fofs; f < total; ++f) sxyz[f - base] = xyz[f];
    }
  }
  lds_async_wait0();
  __syncthreads();

  const int gid = (int)blockIdx.x * BLK + tid;
  if (gid >= npts) return;

  // Contiguous 12-B LDS read (stride-3 words: gcd(3,64)=1, conflict-free).
  float p3[3];
  __builtin_memcpy(p3, &sxyz[tid * 3], 12);
  const float px = p3[0];
  const float py = p3[1];
  const float pz = p3[2];

  // ---- spherical mapping ----
  const float rinv = __builtin_amdgcn_rsqf(px * px + py * py + pz * pz);
  float zn = pz * rinv;
  zn = fminf(1.0f, fmaxf(-1.0f, zn));
  const float phi = asinf(zn);
  float theta = atan2f(py, px);          // atan2(y*s, x*s) == atan2(y, x), s>0

  const float PI_F = 3.14159265358979323846f;
  const float TWO_PI_F = 6.28318530717958647692f;
  const float INV_TWO_PI_F = 0.15915494309189533577f;
  // theta = remainder(theta - u_min, 2*pi) + u_min  (wrap into [-pi, pi))
  float tw = theta + PI_F;
  tw -= floorf(tw * INV_TWO_PI_F) * TWO_PI_F;
  theta = tw - PI_F;

  // k0 = 2/((u_max-u_min) + 2*(inv*PAD - inv/2)) = 1/(pi*(1 + 7/4096)), b0 = 0
  const float K0 = (float)(1.0 / (3.14159265358979323846 * (1.0 + 7.0 / 4096.0)));
  const float K1 = 0.63661977236758134308f;  // 2/pi, b1 = 0
  const float gu = K0 * theta;
  const float gv = K1 * phi;

  // ---- bicubic sample, align_corners=True ----
  // |gu| <= 0.99829 (u padded by 4 cols) -> ix in [3.5, 4099.8]; the 4 x-taps
  // [x0-1, x0+2] are ALWAYS in [2, 4101]: no x clamp/mask needed, so each
  // row of taps is one contiguous 16-B load. Only y can go out of range.
  const float ix = (gu + 1.0f) * (0.5f * (float)(GW - 1));
  const float iy = (gv + 1.0f) * (0.5f * (float)(GH - 1));
  const float xf = floorf(ix);
  const float yf = floorf(iy);
  const float txf = ix - xf;
  const float tyf = iy - yf;
  const int x0 = (int)xf;
  const int y0 = (int)yf;

  float wx[4], wy[4];
  cubic_w(txf, wx);
  cubic_w(tyf, wy);

  // Row base pointers (y clamped) + masked y weights.
  const float* rowp[4];
  float wym[4];
#pragma unroll
  for (int j = 0; j < 4; ++j) {
    const int yi = y0 + j - 1;
    const int yc = min(max(yi, 0), GH - 1);
    wym[j] = (yi >= 0 && yi < GH) ? wy[j] : 0.0f;
    rowp[j] = grid + (size_t)yc * GW + (x0 - 1);
  }

  // Issue all 12 gathers up front so they overlap (one clause, one wait).
  f4v t0[4], t1[4], t2[4];
#pragma unroll
  for (int j = 0; j < 4; ++j) {
    t0[j] = load_f4_unaligned(rowp[j]);
    t1[j] = load_f4_unaligned(rowp[j] + GPLANE);
    t2[j] = load_f4_unaligned(rowp[j] + 2 * GPLANE);
  }

  float acc0 = 0.0f, acc1 = 0.0f, acc2 = 0.0f;
#pragma unroll
  for (int j = 0; j < 4; ++j) {
    const float s0 = fmaf(wx[0], t0[j][0], fmaf(wx[1], t0[j][1],
                     fmaf(wx[2], t0[j][2], wx[3] * t0[j][3])));
    const float s1 = fmaf(wx[0], t1[j][0], fmaf(wx[1], t1[j][1],
                     fmaf(wx[2], t1[j][2], wx[3] * t1[j][3])));
    const float s2 = fmaf(wx[0], t2[j][0], fmaf(wx[1], t2[j][1],
                     fmaf(wx[2], t2[j][2], wx[3] * t2[j][3])));
    acc0 = fmaf(wym[j], s0, acc0);
    acc1 = fmaf(wym[j], s1, acc1);
    acc2 = fmaf(wym[j], s2, acc2);
  }

  // ---- sigmoid + single contiguous 12-B store per lane (b96, coalesced) ----
  float r3[3];
  r3[0] = sigmoid_fast(acc0);
  r3[1] = sigmoid_fast(acc1);
  r3[2] = sigmoid_fast(acc2);
  __builtin_memcpy(out + (size_t)gid * 3, r3, 12);
}

extern "C" void kernel_launch(void* const* d_in, const int* in_sizes, int n_in,
                              void* d_out, int out_size, void* d_ws, size_t ws_size,
                              hipStream_t stream) {
  (void)n_in; (void)out_size; (void)d_ws; (void)ws_size;
  const float* xyz = (const float*)d_in[0];
  const float* grid = (const float*)d_in[1];
  float* out = (float*)d_out;
  const int npts = in_sizes[0] / 3;
  const int nblk = (npts + BLK - 1) / BLK;
  hipLaunchKernelGGL(equ_bicubic_sample, dim3(nblk), dim3(BLK), 0, stream,
                     xyz, grid, out, npts);
}